// SpikeKernelLoss_32341103739073
// MI455X (gfx1250) — compile-verified
//
#include <hip/hip_runtime.h>
#include <hip/hip_bf16.h>
#include <stdint.h>

// SpikeKernelLoss for MI455X (gfx1250), compile-only loop.
//
// outputs: [128, 2048, 100] f32, target: [128, 2048, 100] f32
// loss = 0.5 * sum((outputs - psp(target))^2), psp = leaky integrator over T.
//
// Roofline: 2 x 104.9 MB streamed once, ~160 MFLOP -> memory bound,
// ~9 us floor at 23.3 TB/s. The CDNA5 feature that matters is the async
// global->LDS copy path (ASYNCcnt): each WG's 128-row slab is a fully
// contiguous 51200 B global range per array, so global_load_async_to_lds_b128
// gives perfectly coalesced 16 B/lane traffic with no VGPR round trip; the
// serial T=100 recurrence then runs out of LDS (rows are 400 B = 25x16 B,
// so ds_load_b128 per row, 16B-aligned).

#define T_STEPS      100
#define ROWS_PER_WG  128
#define BLOCK        128
#define SLAB_FLOATS  (ROWS_PER_WG * T_STEPS)   // 12800 floats = 51200 B
#define SLAB_BYTES   (SLAB_FLOATS * 4)
#define COPY_ITERS   (SLAB_BYTES / (BLOCK * 16))  // 25
#define TAU_S_F      5.0f

typedef float float4v __attribute__((ext_vector_type(4)));

__global__ __launch_bounds__(BLOCK)
void spike_loss_stage1(const float* __restrict__ outputs,
                       const float* __restrict__ target,
                       float* __restrict__ partial)
{
    extern __shared__ float lds[];            // 2 * 51200 B = 102400 B dynamic LDS
    float* tgt_s = lds;                       // [ROWS_PER_WG * T_STEPS]
    float* out_s = lds + SLAB_FLOATS;         // [ROWS_PER_WG * T_STEPS]

    const int tid = threadIdx.x;
    const long long base = (long long)blockIdx.x * SLAB_FLOATS;  // contiguous slab

#if defined(__gfx1250__)
    {
        // Hoist all address math: one LDS byte-address (low 32 bits of the
        // generic LDS pointer = DS address) and one global pointer per array,
        // bumped by BLOCK*16 bytes per iteration. Uniform 25-trip loop, all
        // 128 lanes active -> no exec-mask loop, clean codegen.
        unsigned lds_t = (unsigned)(uintptr_t)tgt_s + (unsigned)tid * 16u;
        unsigned lds_o = lds_t + SLAB_BYTES;
        const char* gt = (const char*)(target  + base) + (size_t)tid * 16u;
        const char* go = (const char*)(outputs + base) + (size_t)tid * 16u;
        #pragma unroll 1
        for (int i = 0; i < COPY_ITERS; ++i) {
            asm volatile("global_load_async_to_lds_b128 %0, %1, off"
                         :: "v"(lds_t), "v"(gt) : "memory");
            asm volatile("global_load_async_to_lds_b128 %0, %1, off"
                         :: "v"(lds_o), "v"(go) : "memory");
            lds_t += BLOCK * 16u;
            lds_o += BLOCK * 16u;
            gt    += BLOCK * 16u;
            go    += BLOCK * 16u;
        }
#if __has_builtin(__builtin_amdgcn_s_wait_asynccnt)
        __builtin_amdgcn_s_wait_asynccnt(0);
#else
        asm volatile("s_wait_asynccnt 0" ::: "memory");
#endif
    }
#else
    // Non-gfx1250 fallback: same staging with plain coalesced b128 loads.
    for (int k = 0; k < COPY_ITERS; ++k) {
        int idx = tid + k * BLOCK;
        ((float4v*)tgt_s)[idx] = ((const float4v*)(target  + base))[idx];
        ((float4v*)out_s)[idx] = ((const float4v*)(outputs + base))[idx];
    }
#endif
    __syncthreads();

    // Serial leaky-integrator along T, one row per thread, fed from LDS.
    const float decay   = 1.0f - 1.0f / TAU_S_F;   // 0.8
    const float inv_tau = 1.0f / TAU_S_F;          // 0.2
    float syn = 0.0f;
    float acc = 0.0f;
    const float4v* trow = (const float4v*)(tgt_s + tid * T_STEPS);
    const float4v* orow = (const float4v*)(out_s + tid * T_STEPS);
    #pragma unroll 5
    for (int k = 0; k < T_STEPS / 4; ++k) {
        float4v t4 = trow[k];
        float4v o4 = orow[k];
        float d;
        syn = fmaf(syn, decay, t4.x); d = o4.x - syn * inv_tau; acc = fmaf(d, d, acc);
        syn = fmaf(syn, decay, t4.y); d = o4.y - syn * inv_tau; acc = fmaf(d, d, acc);
        syn = fmaf(syn, decay, t4.z); d = o4.z - syn * inv_tau; acc = fmaf(d, d, acc);
        syn = fmaf(syn, decay, t4.w); d = o4.w - syn * inv_tau; acc = fmaf(d, d, acc);
    }

    // Block reduction (reuse LDS after a barrier).
    __syncthreads();
    tgt_s[tid] = acc;
    __syncthreads();
    for (int s = BLOCK / 2; s > 0; s >>= 1) {
        if (tid < s) tgt_s[tid] += tgt_s[tid + s];
        __syncthreads();
    }
    if (tid == 0) partial[blockIdx.x] = tgt_s[0];
}

__global__ __launch_bounds__(1024)
void spike_loss_stage2(const float* __restrict__ partial, int n,
                       float* __restrict__ out)
{
    __shared__ float red[1024];
    float a = 0.0f;
    for (int i = threadIdx.x; i < n; i += 1024) a += partial[i];
    red[threadIdx.x] = a;
    __syncthreads();
    for (int s = 512; s > 0; s >>= 1) {
        if (threadIdx.x < s) red[threadIdx.x] += red[threadIdx.x + s];
        __syncthreads();
    }
    if (threadIdx.x == 0) out[0] = 0.5f * red[0];   // overwrite: handles poisoned d_out
}

extern "C" void kernel_launch(void* const* d_in, const int* in_sizes, int n_in,
                              void* d_out, int out_size, void* d_ws, size_t ws_size,
                              hipStream_t stream) {
    const float* outputs = (const float*)d_in[0];
    const float* target  = (const float*)d_in[1];
    float* out     = (float*)d_out;
    float* partial = (float*)d_ws;

    const int total = in_sizes[0];                 // 128*2048*100 = 26,214,400
    const int nwg   = total / SLAB_FLOATS;         // 2048 workgroups
    const size_t shmem = 2u * SLAB_FLOATS * sizeof(float);  // 102400 B

    spike_loss_stage1<<<nwg, BLOCK, shmem, stream>>>(outputs, target, partial);
    spike_loss_stage2<<<1, 1024, 0, stream>>>(partial, nwg, out);
}